// GATv2Baseline_17059610100488
// MI455X (gfx1250) — compile-verified
//
#include <hip/hip_runtime.h>
#include <math.h>

// Problem constants (match reference)
#define N_NODES 50000
#define N_EDGES 800000
#define ETOT    (N_EDGES + N_NODES)   // edges + self loops
#define IN_DIM  128
#define HID     64
#define NH      4
#define F       (NH * HID)            // 256
#define MT      (N_NODES / 16)        // 3125 M-tiles (exact)

typedef __attribute__((ext_vector_type(2))) float v2f;
typedef __attribute__((ext_vector_type(8))) float v8f;

// ---------------------------------------------------------------------------
// init: agg[N*F]=0, m[N*NH]=-inf, s[N*NH]=0
__global__ void k_init(float* __restrict__ agg, float* __restrict__ m,
                       float* __restrict__ s) {
  int i = blockIdx.x * blockDim.x + threadIdx.x;
  if (i < N_NODES * F) agg[i] = 0.0f;
  if (i < N_NODES * NH) { m[i] = -3.0e38f; s[i] = 0.0f; }
}

// ---------------------------------------------------------------------------
// WMMA fp32 GEMM: Y[m,n] = sum_k X[m,k]*W[n,k] + B[n], optional ReLU.
// One wave per 16x16 tile; fp32 V_WMMA_F32_16X16X4_F32, K stepped by 4.
// Computes up to two weight sets (Wl/Wr) in one launch.
__global__ void k_gemm_wmma(const float* __restrict__ X, int K,
                            const float* __restrict__ W0, const float* __restrict__ B0,
                            float* __restrict__ Y0,
                            const float* __restrict__ W1, const float* __restrict__ B1,
                            float* __restrict__ Y1,
                            int Ntiles, int nmat, int ldY, int act) {
  int wave = (blockIdx.x * blockDim.x + threadIdx.x) >> 5;   // wave-uniform
  int total = MT * Ntiles * nmat;
  if (wave >= total) return;                                 // whole-wave exit: EXEC stays all-1s

  int tn   = wave % Ntiles;
  int rest = wave / Ntiles;
  int mat  = rest % nmat;
  int tm   = rest / nmat;

  const float* W  = mat ? W1 : W0;
  const float* Bv = mat ? B1 : B0;
  float*       Y  = mat ? Y1 : Y0;

  int lane = threadIdx.x & 31;
  int half = lane >> 4;      // 0: lanes 0-15, 1: lanes 16-31
  int l16  = lane & 15;

  // A layout (f32 16x4): lane=M (l16), VGPR j, half h -> K = k0 + j + 2h
  // B layout (f32 4x16): lane=N (l16), VGPR j, half h -> K = k0 + j + 2h  (B = W^T)
  const float* xrow = X + (size_t)(tm * 16 + l16) * K;
  const float* wrow = W + (size_t)(tn * 16 + l16) * K;

  v8f c = {0.f, 0.f, 0.f, 0.f, 0.f, 0.f, 0.f, 0.f};
  for (int k0 = 0; k0 < K; k0 += 4) {
    v2f a = *(const v2f*)(xrow + k0 + 2 * half);
    v2f b = *(const v2f*)(wrow + k0 + 2 * half);
    c = __builtin_amdgcn_wmma_f32_16x16x4_f32(false, a, false, b, (short)0, c,
                                              false, false);
  }

  // D layout: VGPR r -> row M = tm*16 + r + 8*half, col N = tn*16 + l16
  int col   = tn * 16 + l16;
  float bia = Bv[col];
  int mbase = tm * 16 + 8 * half;
#pragma unroll
  for (int r = 0; r < 8; ++r) {
    float v = c[r] + bia;
    if (act) v = fmaxf(v, 0.0f);
    Y[(size_t)(mbase + r) * ldY + col] = v;
  }
}

// ---------------------------------------------------------------------------
// Edge scoring: one wave per edge. score[e,h] = sum_c lrelu(xl[src]+xr[dst])*att
// Also atomic segment-max into m[dst,h].
__global__ void k_edge_score(const int* __restrict__ ei,
                             const float* __restrict__ xl, const float* __restrict__ xr,
                             const float* __restrict__ att,
                             float* __restrict__ sc, float* __restrict__ m) {
  int e = (blockIdx.x * blockDim.x + threadIdx.x) >> 5;
  if (e >= ETOT) return;
  int lane = threadIdx.x & 31;
  int src = (e < N_EDGES) ? ei[e] : (e - N_EDGES);
  int dst = (e < N_EDGES) ? ei[N_EDGES + e] : (e - N_EDGES);

  const float* pl = xl + (size_t)src * F;
  const float* pr = xr + (size_t)dst * F;

  float acc[NH] = {0.f, 0.f, 0.f, 0.f};
#pragma unroll
  for (int j = 0; j < 8; ++j) {                // channel c = lane + 32j, head = c>>6 = j>>1
    int c = lane + 32 * j;
    float v = pl[c] + pr[c];
    v = (v > 0.f) ? v : 0.2f * v;              // leaky_relu 0.2
    acc[j >> 1] += v * att[c];
  }
#pragma unroll
  for (int h = 0; h < NH; ++h) {
    float a = acc[h];
    for (int off = 16; off > 0; off >>= 1) a += __shfl_xor(a, off, 32);
    acc[h] = a;
  }
  if (lane == 0) {
#pragma unroll
    for (int h = 0; h < NH; ++h) {
      sc[(size_t)e * NH + h] = acc[h];
      atomicMax(&m[(size_t)dst * NH + h], acc[h]);  // global_atomic_max_num_f32
    }
  }
}

// ---------------------------------------------------------------------------
// p = exp(score - m[dst]); segment-sum s[dst,h] += p; store p in place.
__global__ void k_edge_exp(const int* __restrict__ ei, float* __restrict__ sc,
                           const float* __restrict__ m, float* __restrict__ s) {
  int idx = blockIdx.x * blockDim.x + threadIdx.x;
  if (idx >= ETOT * NH) return;
  int e = idx >> 2, h = idx & 3;
  int dst = (e < N_EDGES) ? ei[N_EDGES + e] : (e - N_EDGES);
  float p = __expf(sc[idx] - m[dst * NH + h]);
  sc[idx] = p;
  atomicAdd(&s[dst * NH + h], p);
}

// ---------------------------------------------------------------------------
// agg[dst,h,c] += (p/s[dst,h]) * xl[src,h,c]; one wave per edge.
__global__ void k_edge_agg(const int* __restrict__ ei, const float* __restrict__ xl,
                           const float* __restrict__ sc, const float* __restrict__ s,
                           float* __restrict__ agg) {
  int e = (blockIdx.x * blockDim.x + threadIdx.x) >> 5;
  if (e >= ETOT) return;
  int lane = threadIdx.x & 31;
  int src = (e < N_EDGES) ? ei[e] : (e - N_EDGES);
  int dst = (e < N_EDGES) ? ei[N_EDGES + e] : (e - N_EDGES);

  float alpha[NH];
#pragma unroll
  for (int h = 0; h < NH; ++h)
    alpha[h] = sc[(size_t)e * NH + h] / fmaxf(s[(size_t)dst * NH + h], 1e-16f);

  const float* pl = xl + (size_t)src * F;
  float*       pd = agg + (size_t)dst * F;
#pragma unroll
  for (int j = 0; j < 8; ++j) {
    int c = lane + 32 * j;
    atomicAdd(&pd[c], alpha[j >> 1] * pl[c]);
  }
}

// ---------------------------------------------------------------------------
// head-mean + bias -> LayerNorm(g,beta) -> ELU. One wave per node, 2 ch/lane.
__global__ void k_node_finish(const float* __restrict__ agg, const float* __restrict__ bias,
                              const float* __restrict__ g, const float* __restrict__ beta,
                              float* __restrict__ out) {
  int n = (blockIdx.x * blockDim.x + threadIdx.x) >> 5;
  if (n >= N_NODES) return;
  int lane = threadIdx.x & 31;
  int c0 = lane * 2, c1 = c0 + 1;
  const float* p = agg + (size_t)n * F;

  float v0 = 0.25f * (p[c0] + p[64 + c0] + p[128 + c0] + p[192 + c0]) + bias[c0];
  float v1 = 0.25f * (p[c1] + p[64 + c1] + p[128 + c1] + p[192 + c1]) + bias[c1];

  float sum = v0 + v1;
  for (int off = 16; off > 0; off >>= 1) sum += __shfl_xor(sum, off, 32);
  float mu = sum * (1.0f / 64.0f);
  float d0 = v0 - mu, d1 = v1 - mu;
  float vs = d0 * d0 + d1 * d1;
  for (int off = 16; off > 0; off >>= 1) vs += __shfl_xor(vs, off, 32);
  float inv = rsqrtf(vs * (1.0f / 64.0f) + 1e-5f);

  float y0 = d0 * inv * g[c0] + beta[c0];
  float y1 = d1 * inv * g[c1] + beta[c1];
  y0 = (y0 > 0.f) ? y0 : (__expf(y0) - 1.f);   // ELU
  y1 = (y1 > 0.f) ? y1 : (__expf(y1) - 1.f);
  out[(size_t)n * HID + c0] = y0;
  out[(size_t)n * HID + c1] = y1;
}

// ---------------------------------------------------------------------------
// Final 32 -> 2 projection, one thread per node.
__global__ void k_head2(const float* __restrict__ hm, const float* __restrict__ W,
                        const float* __restrict__ b, float* __restrict__ out) {
  int n = blockIdx.x * blockDim.x + threadIdx.x;
  if (n >= N_NODES) return;
  const float* p = hm + (size_t)n * 32;
  float a0 = b[0], a1 = b[1];
#pragma unroll
  for (int j = 0; j < 32; ++j) {
    float v = p[j];
    a0 += v * W[j];
    a1 += v * W[32 + j];
  }
  out[(size_t)n * 2 + 0] = a0;
  out[(size_t)n * 2 + 1] = a1;
}

// ---------------------------------------------------------------------------
static inline int wave_blocks(long long waves) { return (int)((waves + 7) / 8); }

extern "C" void kernel_launch(void* const* d_in, const int* in_sizes, int n_in,
                              void* d_out, int out_size, void* d_ws, size_t ws_size,
                              hipStream_t stream) {
  const float* x    = (const float*)d_in[0];
  const int*   ei   = (const int*)  d_in[1];
  const float* Wl1  = (const float*)d_in[2];  const float* bl1 = (const float*)d_in[3];
  const float* Wr1  = (const float*)d_in[4];  const float* br1 = (const float*)d_in[5];
  const float* att1 = (const float*)d_in[6];  const float* bias1=(const float*)d_in[7];
  const float* Wl2  = (const float*)d_in[8];  const float* bl2 = (const float*)d_in[9];
  const float* Wr2  = (const float*)d_in[10]; const float* br2 = (const float*)d_in[11];
  const float* att2 = (const float*)d_in[12]; const float* bias2=(const float*)d_in[13];
  const float* g1   = (const float*)d_in[14]; const float* be1 = (const float*)d_in[15];
  const float* g2   = (const float*)d_in[16]; const float* be2 = (const float*)d_in[17];
  const float* Wh1  = (const float*)d_in[18]; const float* bh1 = (const float*)d_in[19];
  const float* Wh2  = (const float*)d_in[20]; const float* bh2 = (const float*)d_in[21];
  float* out = (float*)d_out;

  // workspace layout (floats)
  float* ws  = (float*)d_ws;
  const size_t NF = (size_t)N_NODES * F;       // 12.8M
  float* XL  = ws;
  float* XR  = XL + NF;
  float* AGG = XR + NF;
  float* SC  = AGG + NF;                       // ETOT*NH
  float* M   = SC + (size_t)ETOT * NH;
  float* S   = M  + (size_t)N_NODES * NH;
  float* H1  = S  + (size_t)N_NODES * NH;      // layer outputs [N,64] (reused for layer 2)
  float* HM  = H1 + (size_t)N_NODES * HID;     // MLP mid [N,32]

  const int TB = 256;
  int initBlocks = (int)((NF + TB - 1) / TB);
  int edgeBlocks = wave_blocks(ETOT);
  int nodeBlocks = wave_blocks(N_NODES);
  int expBlocks  = (ETOT * NH + TB - 1) / TB;

  // ---------------- Layer 1 ----------------
  k_init<<<initBlocks, TB, 0, stream>>>(AGG, M, S);
  k_gemm_wmma<<<wave_blocks((long long)MT * 16 * 2), TB, 0, stream>>>(
      x, IN_DIM, Wl1, bl1, XL, Wr1, br1, XR, /*Ntiles=*/16, /*nmat=*/2, F, 0);
  k_edge_score<<<edgeBlocks, TB, 0, stream>>>(ei, XL, XR, att1, SC, M);
  k_edge_exp<<<expBlocks, TB, 0, stream>>>(ei, SC, M, S);
  k_edge_agg<<<edgeBlocks, TB, 0, stream>>>(ei, XL, SC, S, AGG);
  k_node_finish<<<nodeBlocks, TB, 0, stream>>>(AGG, bias1, g1, be1, H1);

  // ---------------- Layer 2 ----------------
  k_init<<<initBlocks, TB, 0, stream>>>(AGG, M, S);
  k_gemm_wmma<<<wave_blocks((long long)MT * 16 * 2), TB, 0, stream>>>(
      H1, HID, Wl2, bl2, XL, Wr2, br2, XR, 16, 2, F, 0);
  k_edge_score<<<edgeBlocks, TB, 0, stream>>>(ei, XL, XR, att2, SC, M);
  k_edge_exp<<<expBlocks, TB, 0, stream>>>(ei, SC, M, S);
  k_edge_agg<<<edgeBlocks, TB, 0, stream>>>(ei, XL, SC, S, AGG);
  k_node_finish<<<nodeBlocks, TB, 0, stream>>>(AGG, bias2, g2, be2, H1);

  // ---------------- MLP head ----------------
  k_gemm_wmma<<<wave_blocks((long long)MT * 2), TB, 0, stream>>>(
      H1, HID, Wh1, bh1, HM, Wh1, bh1, HM, /*Ntiles=*/2, /*nmat=*/1, 32, /*relu=*/1);
  k_head2<<<(N_NODES + TB - 1) / TB, TB, 0, stream>>>(HM, Wh2, bh2, out);
}